// MultiLabelRankingLoss_8469675508229
// MI455X (gfx1250) — compile-verified
//
#include <hip/hip_runtime.h>

// MultiLabelRankingLoss for MI455X (gfx1250, wave32).
// One wave per row: async-copy the 256-float row to LDS (ASYNCcnt path),
// branchless dedupe + hinge accumulation, WMMA f32 16x16x4 cross-lane
// reduction (64 partials -> 16 in one v_wmma), ds_swizzle xor-swaps to
// finish 16 -> 1, two f32 atomics per row, tiny finalize kernel.

typedef __attribute__((ext_vector_type(2))) float v2f;
typedef __attribute__((ext_vector_type(8))) float v8f;
typedef __attribute__((ext_vector_type(4))) int   v4i;

#define AS1 __attribute__((address_space(1)))
#define AS3 __attribute__((address_space(3)))

#define NWAVES   8
#define NTHREADS 256
#define DDIM     256   // developers per row
#define KPOS     8     // positive indices per row
#define MARGIN   1.0f

__global__ void mlrl_init(float* ws) {
    if (threadIdx.x < 2) ws[threadIdx.x] = 0.0f;
}

__global__ __launch_bounds__(NTHREADS)
void mlrl_main(const float* __restrict__ scores,
               const int* __restrict__ pos_idx,
               float* __restrict__ ws,
               int B) {
    __shared__ float lds[NWAVES * DDIM];

    const int lane = threadIdx.x & 31;
    const int wave = threadIdx.x >> 5;
    const int row  = blockIdx.x * NWAVES + wave;
    if (row >= B) return;            // wave-uniform: EXEC stays all-ones below

    const float* grow = scores + (size_t)row * DDIM;
    float*       lrow = &lds[wave * DDIM];

    // ---- Stage this wave's row (1 KB) into LDS: 32 B per lane ----
#if __has_builtin(__builtin_amdgcn_global_load_async_to_lds_b128) && \
    __has_builtin(__builtin_amdgcn_s_wait_asynccnt)
    {
        const float* gsrc = grow + lane * 8;
        float*       ldst = lrow + lane * 8;
        __builtin_amdgcn_global_load_async_to_lds_b128(
            (AS1 v4i*)gsrc, (AS3 v4i*)ldst, 0, 0);
        __builtin_amdgcn_global_load_async_to_lds_b128(
            (AS1 v4i*)(gsrc + 4), (AS3 v4i*)(ldst + 4), 0, 0);
        __builtin_amdgcn_s_wait_asynccnt(0);   // wave-private; no barrier needed
    }
#else
    {
        const float4* g4 = (const float4*)grow;
        float4*       l4 = (float4*)lrow;
        l4[lane * 2 + 0] = g4[lane * 2 + 0];
        l4[lane * 2 + 1] = g4[lane * 2 + 1];
    }
#endif

    // ---- Load positive indices, dedupe (set semantics), gather pos scores ----
    const int* ip = pos_idx + (size_t)row * KPOS;
    int   idx[KPOS];
    float w[KPOS], sp[KPOS];
    int   pcount = 0;
#pragma unroll
    for (int i = 0; i < KPOS; ++i) idx[i] = ip[i];
#pragma unroll
    for (int i = 0; i < KPOS; ++i) {
        bool dup = false;
#pragma unroll
        for (int j = 0; j < KPOS; ++j)
            if (j < i) dup |= (idx[j] == idx[i]);
        w[i]    = dup ? 0.0f : 1.0f;
        pcount += dup ? 0 : 1;
        sp[i]   = lrow[idx[i]];
    }

    // ---- Hinge accumulation: 8 elements per lane, split into 2 accumulators ----
    float accA = 0.0f, accB = 0.0f;
#pragma unroll
    for (int j = 0; j < DDIM / 32; ++j) {
        const int n = lane + j * 32;
        const float s = lrow[n];
        bool isPos = false;
#pragma unroll
        for (int i = 0; i < KPOS; ++i) isPos |= (n == idx[i]);
        float e = 0.0f;
#pragma unroll
        for (int i = 0; i < KPOS; ++i)
            e += w[i] * fmaxf(s - sp[i] + MARGIN, 0.0f);
        e = isPos ? 0.0f : e;          // negatives only
        if (j & 1) accB += e; else accA += e;
    }

    // ---- Wave reduction stage 1 via WMMA: D = ones(16x4) @ B(4x16) ----
    // B layout (f32 4x16): VGPR0 lanes0-15 = B[0][n], lanes16-31 = B[2][n];
    //                      VGPR1 lanes0-15 = B[1][n], lanes16-31 = B[3][n].
    // => D[m][n] = accA(n) + accB(n) + accA(n+16) + accB(n+16) for all m,
    // so d[0] holds colsum[lane & 15] in every lane: 64 partials -> 16.
    v2f amat; amat[0] = 1.0f; amat[1] = 1.0f;
    v2f bmat; bmat[0] = accA; bmat[1] = accB;
    v8f cmat = {};
    cmat = __builtin_amdgcn_wmma_f32_16x16x4_f32(
        /*neg_a=*/false, amat, /*neg_b=*/false, bmat,
        /*c_mod=*/(short)0, cmat, /*reuse_a=*/false, /*reuse_b=*/false);

    float total = cmat[0];
    // ---- stage 2: 16 -> 1. Values are 16-periodic across the wave, so
    // plain xor-swaps by 1/2/4/8 suffice; ds_swizzle BITMASK_PERM
    // (and=0x1F, or=0, xor=k -> offset 0x7C0k) does each in one DS op. ----
#if __has_builtin(__builtin_amdgcn_ds_swizzle)
    total += __int_as_float(__builtin_amdgcn_ds_swizzle(__float_as_int(total), 0x7C01));
    total += __int_as_float(__builtin_amdgcn_ds_swizzle(__float_as_int(total), 0x7C02));
    total += __int_as_float(__builtin_amdgcn_ds_swizzle(__float_as_int(total), 0x7C04));
    total += __int_as_float(__builtin_amdgcn_ds_swizzle(__float_as_int(total), 0x7C08));
#else
    total += __shfl_xor(total, 1);
    total += __shfl_xor(total, 2);
    total += __shfl_xor(total, 4);
    total += __shfl_xor(total, 8);
#endif

    if (lane == 0) {
        atomicAdd(&ws[0], total);
        atomicAdd(&ws[1], (float)((DDIM - pcount) * pcount));
    }
}

__global__ void mlrl_finalize(const float* __restrict__ ws, float* __restrict__ out) {
    const float pairs = ws[1];
    out[0] = (pairs > 0.0f) ? (ws[0] / fmaxf(pairs, 1.0f)) : 0.0f;
}

extern "C" void kernel_launch(void* const* d_in, const int* in_sizes, int n_in,
                              void* d_out, int out_size, void* d_ws, size_t ws_size,
                              hipStream_t stream) {
    const float* scores = (const float*)d_in[0];
    const int*   pos    = (const int*)d_in[1];   // harness delivers integer inputs as int32
    float*       ws     = (float*)d_ws;
    float*       out    = (float*)d_out;

    const int B = in_sizes[1] / KPOS;            // 2048
    const int blocks = (B + NWAVES - 1) / NWAVES;

    mlrl_init<<<1, 32, 0, stream>>>(ws);
    mlrl_main<<<blocks, NTHREADS, 0, stream>>>(scores, pos, ws, B);
    mlrl_finalize<<<1, 1, 0, stream>>>(ws, out);
}